// E_01_HSE_49924699848911
// MI455X (gfx1250) — compile-verified
//
#include <hip/hip_runtime.h>

typedef float v2f __attribute__((ext_vector_type(2)));
typedef float v8f __attribute__((ext_vector_type(8)));

// Problem constants (from reference)
constexpr int BATCH = 32;
constexpr int LSEQ  = 4096;
constexpr int CCH   = 64;
constexpr int PL    = 64;
constexpr int PC    = 16;
constexpr int NP    = 256;
constexpr int NOUT  = 512;
constexpr int DIN   = PL * (PC * 2);   // 2048
constexpr float TSTEP = 1.0f / 1000.0f;

// Tiling
constexpr int MT  = 32;          // rows per block
constexpr int NT  = 64;          // cols per N-group (per 4-wave column set)
constexpr int NB  = NOUT / NT;   // 8 N-groups, all owned by one block in kernel 1

constexpr int KC1 = 16;          // K chunk kernel 1 (chunk-uniform patch/time split)
constexpr int LD1 = KC1 + 4;     // 20-dword padded stride: 16B-aligned, conflict-free

constexpr int KC2 = 64;          // K chunk kernel 2
constexpr int LD2 = KC2 + 4;     // 68-dword padded stride

// ---------------------------------------------------------------------------
// Kernel 1: gather feats tile once -> GEMM1 over ALL 512 cols -> +b1 -> SiLU
//   h is (8192 x 512) row-major in workspace; row = b*NP + p
//   Block: 256 threads (8 waves, 2 M x 4 N), 8 accumulators per wave (N-groups)
// ---------------------------------------------------------------------------
__global__ __launch_bounds__(256)
void E_01_HSE_k1(const float* __restrict__ x,
                 const int*   __restrict__ startL,
                 const int*   __restrict__ startC,
                 const float* __restrict__ w1,
                 const float* __restrict__ b1,
                 float*       __restrict__ h)
{
    __shared__ __align__(16) float As[MT * LD1];       //  2.5 KB
    __shared__ __align__(16) float Bs[NOUT * LD1];     // 40.0 KB

    const int tid  = threadIdx.x;
    const int lane = tid & 31;
    const int wave = tid >> 5;
    const int wm   = wave & 1;    // wave M row (0..1)
    const int wn   = wave >> 1;   // wave N col within a 64-group (0..3)
    const int ln   = lane & 15;
    const int hi   = lane >> 4;   // half-wave: K pair select / M half select

    const int row0 = blockIdx.x * MT;
    const int bb   = row0 >> 8;              // batch is uniform per block (32 | 256)

    // This thread gathers element column kk of rows r0 and r0+16.
    const int r0 = tid >> 4;                 // 0..15
    const int kk = tid & 15;                 // 0..15
    int sl[2], sc[2];
#pragma unroll
    for (int e = 0; e < 2; ++e) {
        const int row = row0 + r0 + e * 16;
        const int p   = row & (NP - 1);
        sl[e] = startL[bb * NP + p];
        sc[e] = startC[bb * NP + p];
    }

    v8f acc[NB] = {};

    for (int kc = 0; kc < DIN; kc += KC1) {
        // ---- gather A tile (32 x 16). Chunk kind is uniform: kc%32==0 -> patch
        //      half (x gather), kc%32==16 -> time half (computed, no loads).
        const int  lofs    = kc >> 5;
        const bool isPatch = ((kc & 16) == 0);
#pragma unroll
        for (int e = 0; e < 2; ++e) {
            const int l = (sl[e] + lofs) & (LSEQ - 1);
            float v;
            if (isPatch) {
                const int cc = (sc[e] + kk) & (CCH - 1);
                v = x[((size_t)bb * LSEQ + l) * CCH + cc];
            } else {
                v = (float)l * TSTEP;        // time channel, per-row constant
            }
            As[(r0 + e * 16) * LD1 + kk] = v;
        }
        // ---- B tile: Bs[c][k] = w1[c][kc+k] for all 512 cols, float4 loads
#pragma unroll
        for (int e = 0; e < (NOUT * KC1) / (256 * 4); ++e) {   // 8
            const int idx = e * 256 + tid;
            const int cc  = idx >> 2;                 // 0..511
            const int kq  = (idx & 3) << 2;           // 0,4,8,12
            const float4 v = *(const float4*)&w1[(size_t)cc * DIN + kc + kq];
            *(float4*)&Bs[cc * LD1 + kq] = v;
        }
        __syncthreads();

        // ---- MMA: 4 k-steps x 8 independent accumulator chains
        const float* Ab = &As[(wm * 16 + ln) * LD1 + hi * 2];
#pragma unroll
        for (int i = 0; i < KC1 / 4; ++i) {
            const v2f a = *(const v2f*)(Ab + i * 4);
#pragma unroll
            for (int nb = 0; nb < NB; ++nb) {
                const v2f b = *(const v2f*)&Bs[(nb * NT + wn * 16 + ln) * LD1 + hi * 2 + i * 4];
                acc[nb] = __builtin_amdgcn_wmma_f32_16x16x4_f32(
                              false, a, false, b, (short)0, acc[nb], false, false);
            }
        }
        __syncthreads();
    }

    // ---- epilogue: +bias, SiLU, store
#pragma unroll
    for (int nb = 0; nb < NB; ++nb) {
        const int col  = nb * NT + wn * 16 + ln;
        const float bv = b1[col];
#pragma unroll
        for (int r = 0; r < 8; ++r) {
            const int row = row0 + wm * 16 + hi * 8 + r;
            float v = acc[nb][r] + bv;
            v = v / (1.0f + __expf(-v));               // silu
            h[(size_t)row * NOUT + col] = v;
        }
    }
}

// ---------------------------------------------------------------------------
// Kernel 2: out = h @ w2^T + b2    (M=8192, K=512, N=512)
// ---------------------------------------------------------------------------
__global__ __launch_bounds__(256)
void E_01_HSE_k2(const float* __restrict__ hbuf,
                 const float* __restrict__ w2,
                 const float* __restrict__ b2,
                 float*       __restrict__ out)
{
    __shared__ __align__(16) float As[MT * LD2];
    __shared__ __align__(16) float Bs[NT * LD2];

    const int tid  = threadIdx.x;
    const int lane = tid & 31;
    const int wave = tid >> 5;
    const int wm   = wave & 1;
    const int wn   = wave >> 1;
    const int ln   = lane & 15;
    const int hi   = lane >> 4;

    const int row0 = blockIdx.x * MT;
    const int col0 = blockIdx.y * NT;

    v8f acc = {};

    for (int kc = 0; kc < NOUT; kc += KC2) {
        // A tile from h: 32x64, float4 loads (2 per thread)
#pragma unroll
        for (int e = 0; e < (MT * KC2) / (256 * 4); ++e) {
            const int idx = e * 256 + tid;
            const int r   = idx >> 4;                // 0..31
            const int kq  = (idx & 15) << 2;
            const float4 v = *(const float4*)&hbuf[(size_t)(row0 + r) * NOUT + kc + kq];
            *(float4*)&As[r * LD2 + kq] = v;
        }
        // B tile from w2: Bs[c][k] = w2[col0+c][kc+k]
#pragma unroll
        for (int e = 0; e < (NT * KC2) / (256 * 4); ++e) {
            const int idx = e * 256 + tid;
            const int cc  = idx >> 4;
            const int kq  = (idx & 15) << 2;
            const float4 v = *(const float4*)&w2[(size_t)(col0 + cc) * NOUT + kc + kq];
            *(float4*)&Bs[cc * LD2 + kq] = v;
        }
        __syncthreads();

        const float* Ab = &As[(wm * 16 + ln) * LD2 + hi * 2];
        const float* Bb = &Bs[(wn * 16 + ln) * LD2 + hi * 2];
#pragma unroll
        for (int i = 0; i < KC2 / 4; ++i) {
            v2f a = *(const v2f*)(Ab + i * 4);
            v2f b = *(const v2f*)(Bb + i * 4);
            acc = __builtin_amdgcn_wmma_f32_16x16x4_f32(
                      false, a, false, b, (short)0, acc, false, false);
        }
        __syncthreads();
    }

    const int col  = col0 + wn * 16 + ln;
    const float bv = b2[col];
#pragma unroll
    for (int r = 0; r < 8; ++r) {
        const int row = row0 + wm * 16 + hi * 8 + r;
        out[(size_t)row * NOUT + col] = acc[r] + bv;
    }
}

// ---------------------------------------------------------------------------
extern "C" void kernel_launch(void* const* d_in, const int* in_sizes, int n_in,
                              void* d_out, int out_size, void* d_ws, size_t ws_size,
                              hipStream_t stream)
{
    const float* x      = (const float*)d_in[0];
    const int*   startL = (const int*)  d_in[1];
    const int*   startC = (const int*)  d_in[2];
    const float* w1     = (const float*)d_in[3];
    const float* b1     = (const float*)d_in[4];
    const float* w2     = (const float*)d_in[5];
    const float* b2     = (const float*)d_in[6];
    float*       out    = (float*)d_out;
    float*       h      = (float*)d_ws;   // 8192 * 512 * 4B = 16 MB scratch

    const dim3 block(256);
    const dim3 grid1((BATCH * NP) / MT);             // 256 blocks, full N per block
    const dim3 grid2((BATCH * NP) / MT, NOUT / NT);  // (256, 8)

    hipLaunchKernelGGL(E_01_HSE_k1, grid1, block, 0, stream,
                       x, startL, startC, w1, b1, h);
    hipLaunchKernelGGL(E_01_HSE_k2, grid2, block, 0, stream,
                       h, w2, b2, out);
}